// GATBasedMolecularGraphNeuralNetworkAlt_54872502173937
// MI455X (gfx1250) — compile-verified
//
#include <hip/hip_runtime.h>
#include <hip/hip_bf16.h>
#include <float.h>

// ---------------- problem constants (from reference) ----------------
#define NN 100000   // nodes
#define EE 400000   // edges
#define GG 2000     // graphs
#define IN_CH 48
#define HID 128
#define NLAYERS 6
#define BN_EPS 1e-5f
#define SLOPE 0.01f

#define ROW_TILES 5           // 5 x 16 = 80 rows per block; 100000 = 1250 * 80 exactly

typedef __attribute__((ext_vector_type(2))) float v2f;
typedef __attribute__((ext_vector_type(8))) float v8f;

// ---------------- small helpers ----------------
__device__ __forceinline__ float lrelu(float x) { return x >= 0.f ? x : SLOPE * x; }

// order-preserving float<->uint encoding for atomic max
__device__ __forceinline__ unsigned fenc(float x) {
    unsigned b = __float_as_uint(x);
    return (b & 0x80000000u) ? ~b : (b | 0x80000000u);
}
__device__ __forceinline__ float fdec(unsigned k) {
    unsigned b = (k & 0x80000000u) ? (k & 0x7FFFFFFFu) : ~k;
    return __uint_as_float(b);
}

// ---------------- generic elementwise kernels ----------------
__global__ void zero_f32_kernel(float* p, size_t n) {
    for (size_t i = (size_t)blockIdx.x * blockDim.x + threadIdx.x; i < n;
         i += (size_t)gridDim.x * blockDim.x)
        p[i] = 0.f;
}

__global__ void copy_f32_kernel(const float* __restrict__ src, float* __restrict__ dst, size_t n) {
    for (size_t i = (size_t)blockIdx.x * blockDim.x + threadIdx.x; i < n;
         i += (size_t)gridDim.x * blockDim.x)
        dst[i] = src[i];
}

// ---------------- GIN edge aggregation: z[dst] += h[src] ----------------
template <int HH>
__global__ void scatter_add_kernel(const float* __restrict__ h, const int* __restrict__ src,
                                   const int* __restrict__ dst, float* __restrict__ z, int nE) {
    const int per = blockDim.x / HH;                // edges per block
    const int e   = blockIdx.x * per + threadIdx.x / HH;
    const int f   = threadIdx.x % HH;
    if (e < nE) {
        atomicAdd(&z[(size_t)dst[e] * HH + f], h[(size_t)src[e] * HH + f]);
    }
}

// ---------------- fp32 WMMA GEMM:  Y[N x 128] = A[N x K] @ W[K x 128] + bias ----------
// Branch-free, K compile-time. Grid: N/(16*ROW_TILES) blocks, 256 threads (8 waves).
// Wave w owns the 16-col tile [16w, 16w+16); each wave computes ROW_TILES 16x16 C tiles,
// reusing one B fragment per K-step across all row tiles (5 independent WMMA chains).
template <int K>
__global__ void gemm_wmma_kernel(const float* __restrict__ A, const float* __restrict__ W,
                                 const float* __restrict__ bias, float* __restrict__ Y) {
    const int lane  = threadIdx.x & 31;
    const int wave  = threadIdx.x >> 5;          // 0..7 -> 16-col tile
    const int row0  = blockIdx.x * (16 * ROW_TILES);
    const int m     = lane & 15;                 // A-row within tile / B,C column
    const int khalf = lane >> 4;                 // 0 or 1 (K-pair selector)
    const int col   = wave * 16 + m;

    // bias folded into accumulator init (bias is column-uniform per lane in C/D layout)
    const float bv = bias[col];
    v8f acc[ROW_TILES];
#pragma unroll
    for (int r = 0; r < ROW_TILES; ++r) {
#pragma unroll
        for (int v = 0; v < 8; ++v) acc[r][v] = bv;
    }

    for (int k0 = 0; k0 < K; k0 += 4) {
        const int ka = k0 + khalf * 2;
        // B fragment (4x16 f32): lane holds W[ka][col], W[ka+1][col]  -- shared by all row tiles
        v2f b = {W[(size_t)ka * HID + col], W[(size_t)(ka + 1) * HID + col]};
#pragma unroll
        for (int r = 0; r < ROW_TILES; ++r) {
            // A fragment (16x4 f32): lane holds A[row][ka], A[row][ka+1]
            const float* ap = A + (size_t)(row0 + r * 16 + m) * K + ka;
            v2f a = {ap[0], ap[1]};
            acc[r] = __builtin_amdgcn_wmma_f32_16x16x4_f32(false, a, false, b, (short)0,
                                                           acc[r], false, false);
        }
    }

    // C/D layout: VGPR v -> row (v + 8*khalf) within tile, col = lane%16
#pragma unroll
    for (int r = 0; r < ROW_TILES; ++r) {
#pragma unroll
        for (int v = 0; v < 8; ++v) {
            Y[(size_t)(row0 + r * 16 + v + 8 * khalf) * HID + col] = acc[r][v];
        }
    }
}

// ---------------- BatchNorm statistics (two pass, atomic partials) ----------------
__global__ void bnstats_partial_kernel(const float* __restrict__ X, int n,
                                       float* __restrict__ sum, float* __restrict__ sumsq) {
    const int f = threadIdx.x;   // blockDim.x == 128
    float s = 0.f, ss = 0.f;
    for (int r = blockIdx.x; r < n; r += gridDim.x) {
        float v = X[(size_t)r * HID + f];
        s += v;
        ss += v * v;
    }
    atomicAdd(&sum[f], s);
    atomicAdd(&sumsq[f], ss);
}

// finalize into per-feature affine form: y = scale*x + shift
__global__ void bnstats_final_kernel(const float* __restrict__ sum, const float* __restrict__ sumsq,
                                     float invN,
                                     const float* __restrict__ gamma, const float* __restrict__ beta,
                                     float* __restrict__ scale, float* __restrict__ shift) {
    const int f = threadIdx.x;   // one block of 128
    float mval = sum[f] * invN;
    float vval = sumsq[f] * invN - mval * mval;
    float s = gamma[f] * rsqrtf(vval + BN_EPS);
    scale[f] = s;
    shift[f] = beta[f] - mval * s;
}

// Y = lrelu(scale*X + shift)   (Y may alias X)
__global__ void bn_lrelu_kernel(const float* __restrict__ X, float* __restrict__ Y, size_t n,
                                const float* __restrict__ scale, const float* __restrict__ shift) {
    for (size_t i = (size_t)blockIdx.x * blockDim.x + threadIdx.x; i < n;
         i += (size_t)gridDim.x * blockDim.x) {
        const int f = (int)(i & (HID - 1));
        Y[i] = lrelu(fmaf(scale[f], X[i], shift[f]));
    }
}

// ---------------- pooling ----------------
__global__ void pool_init_kernel(float* __restrict__ psum, unsigned* __restrict__ pmax,
                                 float* __restrict__ cnt) {
    const unsigned NEG_ENC = fenc(-FLT_MAX);
    for (size_t i = (size_t)blockIdx.x * blockDim.x + threadIdx.x; i < (size_t)GG * HID;
         i += (size_t)gridDim.x * blockDim.x) {
        psum[i] = 0.f;
        pmax[i] = NEG_ENC;
        if (i < GG) cnt[i] = 0.f;
    }
}

__global__ void pool_count_kernel(const int* __restrict__ batch, float* __restrict__ cnt) {
    for (size_t i = (size_t)blockIdx.x * blockDim.x + threadIdx.x; i < NN;
         i += (size_t)gridDim.x * blockDim.x)
        atomicAdd(&cnt[batch[i]], 1.0f);
}

__global__ void pool_accum_kernel(const float* __restrict__ h, const int* __restrict__ batch,
                                  float* __restrict__ psum, unsigned* __restrict__ pmax) {
    for (size_t i = (size_t)blockIdx.x * blockDim.x + threadIdx.x; i < (size_t)NN * HID;
         i += (size_t)gridDim.x * blockDim.x) {
        const int n = (int)(i >> 7);
        const int f = (int)(i & (HID - 1));
        const int g = batch[n];
        const float v = h[i];
        atomicAdd(&psum[(size_t)g * HID + f], v);
        atomicMax(&pmax[(size_t)g * HID + f], fenc(v));
    }
}

// emb[g][0..127] = mean pool, emb[g][128..255] = max pool
__global__ void pool_final_kernel(const float* __restrict__ psum, const unsigned* __restrict__ pmax,
                                  const float* __restrict__ cnt, float* __restrict__ emb) {
    for (size_t i = (size_t)blockIdx.x * blockDim.x + threadIdx.x; i < (size_t)GG * HID;
         i += (size_t)gridDim.x * blockDim.x) {
        const int g = (int)(i >> 7);
        const int f = (int)(i & (HID - 1));
        emb[(size_t)g * 256 + f]       = psum[i] / fmaxf(cnt[g], 1.0f);
        emb[(size_t)g * 256 + 128 + f] = fdec(pmax[i]);
    }
}

// ---------------- final MLP (tiny: G x 256 -> 64 -> 3) ----------------
__global__ void fc1_kernel(const float* __restrict__ emb, const float* __restrict__ w,
                           const float* __restrict__ b, float* __restrict__ hid) {
    for (int i = blockIdx.x * blockDim.x + threadIdx.x; i < GG * 64;
         i += gridDim.x * blockDim.x) {
        const int g = i >> 6, j = i & 63;
        float s = b[j];
        const float* eg = emb + (size_t)g * 256;
        for (int k = 0; k < 256; ++k) s += eg[k] * w[k * 64 + j];
        hid[i] = lrelu(s);
    }
}

__global__ void fc2_kernel(const float* __restrict__ hid, const float* __restrict__ w,
                           const float* __restrict__ b, float* __restrict__ out) {
    for (int i = blockIdx.x * blockDim.x + threadIdx.x; i < GG * 3;
         i += gridDim.x * blockDim.x) {
        const int g = i / 3, c = i % 3;
        float s = b[c];
        const float* hg = hid + (size_t)g * 64;
        for (int k = 0; k < 64; ++k) s += hg[k] * w[k * 3 + c];
        out[i] = s;
    }
}

// ---------------- host orchestration ----------------
extern "C" void kernel_launch(void* const* d_in, const int* in_sizes, int n_in,
                              void* d_out, int out_size, void* d_ws, size_t ws_size,
                              hipStream_t stream) {
    (void)in_sizes; (void)n_in; (void)out_size; (void)ws_size;

    const float* x     = (const float*)d_in[0];   // [N,48]
    // d_in[1] edge_attr: unused by GINConv
    const float* W1a   = (const float*)d_in[2];   // [48,128]
    const float* W1b   = (const float*)d_in[3];   // [5,128,128]
    const float* b1    = (const float*)d_in[4];   // [6,128]
    const float* g1    = (const float*)d_in[5];   // [6,128]
    const float* be1   = (const float*)d_in[6];   // [6,128]
    const float* W2    = (const float*)d_in[7];   // [6,128,128]
    const float* b2    = (const float*)d_in[8];   // [6,128]
    const float* gn    = (const float*)d_in[9];   // [5,128]
    const float* bn_b  = (const float*)d_in[10];  // [5,128]
    const float* fc_w  = (const float*)d_in[11];  // [256,64]
    const float* fc_b  = (const float*)d_in[12];  // [64]
    const float* fc2_w = (const float*)d_in[13];  // [64,3]
    const float* fc2_b = (const float*)d_in[14];  // [3]
    const int*   eidx  = (const int*)d_in[15];    // [2,E]
    const int*   batch = (const int*)d_in[16];    // [N]

    const int* src = eidx;
    const int* dst = eidx + EE;

    // workspace layout (floats)
    float* bufA   = (float*)d_ws;                    // [N,128]
    float* bufB   = bufA + (size_t)NN * HID;         // [N,128]
    float* bufC   = bufB + (size_t)NN * HID;         // [N,128]
    float* ssum   = bufC + (size_t)NN * HID;         // [128]
    float* ssq    = ssum + HID;                      // [128]
    float* scale1 = ssq + HID;                       // [128]
    float* shift1 = scale1 + HID;                    // [128]
    float* scale2 = shift1 + HID;                    // [128]
    float* shift2 = scale2 + HID;                    // [128]
    float* psum   = shift2 + HID;                    // [G,128]
    unsigned* pmax = (unsigned*)(psum + (size_t)GG * HID); // [G,128]
    float* cnt    = (float*)(pmax + (size_t)GG * HID);     // [G]
    float* hid    = cnt + GG;                        // [G,64]

    float* outp = (float*)d_out;         // [G,3]
    float* emb  = outp + (size_t)GG * 3; // [G,256]

    const float invN = 1.0f / (float)NN;
    const int gemmGrid = NN / (16 * ROW_TILES);  // 1250, exact

    for (int l = 0; l < NLAYERS; ++l) {
        const int    Kin = (l == 0) ? IN_CH : HID;
        const float* h   = (l == 0) ? x : bufA;
        const float* W1  = (l == 0) ? W1a : (W1b + (size_t)(l - 1) * HID * HID);

        // z = h  (then scatter-add neighbors)
        const size_t nz = (size_t)NN * Kin;
        copy_f32_kernel<<<2048, 256, 0, stream>>>(h, bufB, nz);
        if (Kin == IN_CH) {
            scatter_add_kernel<IN_CH><<<(EE + 3) / 4, 192, 0, stream>>>(h, src, dst, bufB, EE);
        } else {
            scatter_add_kernel<HID><<<(EE + 1) / 2, 256, 0, stream>>>(h, src, dst, bufB, EE);
        }

        // t1 = z @ W1 + b1[l]
        if (Kin == IN_CH) {
            gemm_wmma_kernel<IN_CH><<<gemmGrid, 256, 0, stream>>>(bufB, W1, b1 + l * HID, bufC);
        } else {
            gemm_wmma_kernel<HID><<<gemmGrid, 256, 0, stream>>>(bufB, W1, b1 + l * HID, bufC);
        }

        // inner BN stats of t1 -> affine scale/shift, then t1' = lrelu(bn(t1)) into bufB
        zero_f32_kernel<<<1, 256, 0, stream>>>(ssum, 2 * HID);
        bnstats_partial_kernel<<<960, HID, 0, stream>>>(bufC, NN, ssum, ssq);
        bnstats_final_kernel<<<1, HID, 0, stream>>>(ssum, ssq, invN, g1 + l * HID, be1 + l * HID,
                                                    scale1, shift1);
        bn_lrelu_kernel<<<2048, 256, 0, stream>>>(bufC, bufB, (size_t)NN * HID, scale1, shift1);

        // h_new = t1' @ W2[l] + b2[l]
        gemm_wmma_kernel<HID><<<gemmGrid, 256, 0, stream>>>(bufB, W2 + (size_t)l * HID * HID,
                                                            b2 + l * HID, bufA);

        if (l < NLAYERS - 1) {
            // outer BN stats + in-place bn+lrelu
            zero_f32_kernel<<<1, 256, 0, stream>>>(ssum, 2 * HID);
            bnstats_partial_kernel<<<960, HID, 0, stream>>>(bufA, NN, ssum, ssq);
            bnstats_final_kernel<<<1, HID, 0, stream>>>(ssum, ssq, invN, gn + l * HID,
                                                        bn_b + l * HID, scale2, shift2);
            bn_lrelu_kernel<<<2048, 256, 0, stream>>>(bufA, bufA, (size_t)NN * HID, scale2, shift2);
        }
    }

    // ---- dual global pooling ----
    pool_init_kernel<<<512, 256, 0, stream>>>(psum, pmax, cnt);
    pool_count_kernel<<<512, 256, 0, stream>>>(batch, cnt);
    pool_accum_kernel<<<2048, 256, 0, stream>>>(bufA, batch, psum, pmax);
    pool_final_kernel<<<512, 256, 0, stream>>>(psum, pmax, cnt, emb);

    // ---- head MLP ----
    fc1_kernel<<<(GG * 64 + 255) / 256, 256, 0, stream>>>(emb, fc_w, fc_b, hid);
    fc2_kernel<<<(GG * 3 + 255) / 256, 256, 0, stream>>>(hid, fc2_w, fc2_b, outp);
}